// ExpertFFN_68384469286912
// MI455X (gfx1250) — compile-verified
//
#include <hip/hip_runtime.h>
#include <hip/hip_bf16.h>
#include <math.h>

// ---------------------------------------------------------------------------
// MoE expert-FFN (summed-dispatch variant) for MI455X / gfx1250.
// Memory-bound: W1+W2 = 1 GB fp32 streamed once (~43us @ 23.3 TB/s) dominates.
// fp32 WMMA (V_WMMA_F32_16X16X4_F32) for all matrix math (E*K == 16 == M), and
// CDNA5 async LDS copies (ASYNCcnt / s_wait_asynccnt) double-buffer the
// weight stream in the two FFN kernels.
// ---------------------------------------------------------------------------

#define NB   4
#define SS   2048
#define NTOK (NB * SS)      // 8192 tokens
#define HH   2048
#define FF   8192
#define NE   8
#define KK   2
#define NSLOT (NE * KK)     // 16 == WMMA M

typedef __attribute__((ext_vector_type(2))) float v2f;
typedef __attribute__((ext_vector_type(8))) float v8f;
typedef __attribute__((ext_vector_type(4))) int   v4i;

static __device__ __forceinline__ v8f wmma_f32_k4(v2f a, v2f b, v8f c) {
    // D = A(16x4 f32) * B(4x16 f32) + C(16x16 f32)
    return __builtin_amdgcn_wmma_f32_16x16x4_f32(
        /*neg_a=*/false, a, /*neg_b=*/false, b,
        /*c_mod=*/(short)0, c, /*reuse_a=*/false, /*reuse_b=*/false);
}

static __device__ __forceinline__ float gelu_exact(float v) {
    return 0.5f * v * (1.0f + erff(v * 0.70710678118654752f));
}

// ---- CDNA5 async global->LDS copy (16B per lane, 512B per instruction) ----
#if __has_builtin(__builtin_amdgcn_global_load_async_to_lds_b128)
#define HAVE_ASYNC_LDS 1
#else
#define HAVE_ASYNC_LDS 0
#endif

#define AS1 __attribute__((address_space(1)))
#define AS3 __attribute__((address_space(3)))

static __device__ __forceinline__ void stage16(const float* g, float* l) {
#if HAVE_ASYNC_LDS
    __builtin_amdgcn_global_load_async_to_lds_b128(
        (AS1 v4i*)(AS1 void*)(void*)g,
        (AS3 v4i*)(AS3 void*)(void*)l, 0, 0);
#else
    // synchronous fallback (still 16B vector moves)
    const float4 v = *(const float4*)g;
    *(float4*)l = v;
#endif
}

static __device__ __forceinline__ void wait_async_8() {
#if HAVE_ASYNC_LDS && __has_builtin(__builtin_amdgcn_s_wait_asynccnt)
    __builtin_amdgcn_s_wait_asynccnt(8);
#else
    asm volatile("s_wait_asynccnt 0x8" ::: "memory");
#endif
}
static __device__ __forceinline__ void wait_async_0() {
#if HAVE_ASYNC_LDS && __has_builtin(__builtin_amdgcn_s_wait_asynccnt)
    __builtin_amdgcn_s_wait_asynccnt(0);
#else
    asm volatile("s_wait_asynccnt 0x0" ::: "memory");
#endif
}

// ---------------------------------------------------------------------------
// K1: router GEMM + softmax + top-2.  One wave per token, coalesced x reads,
// wave32 shuffle reduction, Wr (64KB) L2-resident.
// ---------------------------------------------------------------------------
__global__ __launch_bounds__(256)
void k_router(const float* __restrict__ x, const float* __restrict__ Wr,
              const float* __restrict__ br, float* __restrict__ G,
              int* __restrict__ idxb) {
    const int wave = threadIdx.x >> 5;
    const int lane = threadIdx.x & 31;
    const int n = blockIdx.x * 8 + wave;
    const float* xr = x + (size_t)n * HH;

    float acc[NE];
#pragma unroll
    for (int e = 0; e < NE; ++e) acc[e] = 0.0f;

    for (int h = lane; h < HH; h += 32) {
        const float xv = xr[h];
        const float* wr = Wr + (size_t)h * NE;
#pragma unroll
        for (int e = 0; e < NE; ++e) acc[e] = fmaf(xv, wr[e], acc[e]);
    }
#pragma unroll
    for (int e = 0; e < NE; ++e) {
#pragma unroll
        for (int off = 16; off > 0; off >>= 1)
            acc[e] += __shfl_xor(acc[e], off, 32);
    }
    if (lane == 0) {
        float l[NE], g[NE];
        float m = -1e30f;
#pragma unroll
        for (int e = 0; e < NE; ++e) { l[e] = acc[e] + br[e]; m = fmaxf(m, l[e]); }
        float s = 0.0f;
#pragma unroll
        for (int e = 0; e < NE; ++e) { g[e] = __expf(l[e] - m); s += g[e]; }
        const float inv = 1.0f / s;
        int i1 = 0;
#pragma unroll
        for (int e = 1; e < NE; ++e) if (g[e] > g[i1]) i1 = e;
        int i2 = (i1 == 0) ? 1 : 0;
#pragma unroll
        for (int e = 0; e < NE; ++e) if (e != i1 && g[e] > g[i2]) i2 = e;
        G[n * KK + 0] = g[i1] * inv;
        G[n * KK + 1] = g[i2] * inv;
        idxb[n * KK + 0] = i1;
        idxb[n * KK + 1] = i2;
    }
}

// ---------------------------------------------------------------------------
// K2: init scratch: tok = 0, midacc = 0, Eo = broadcast(b2).
// ---------------------------------------------------------------------------
__global__ __launch_bounds__(256)
void k_init(float* __restrict__ tok, float* __restrict__ midacc,
            float* __restrict__ Eo, const float* __restrict__ b2) {
    const int i = blockIdx.x * 256 + threadIdx.x;       // [0, 16*FF)
    midacc[i] = 0.0f;
    if (i < NSLOT * HH) {
        tok[i] = 0.0f;
        const int r = i >> 11;            // slot = e*K + k
        const int h = i & (HH - 1);
        Eo[i] = b2[(r >> 1) * HH + h];
    }
}

// ---------------------------------------------------------------------------
// K3: dispatch as WMMA GEMM:  tok(16 x H) = P^T(16 x N) * X(N x H).
// One wave per (16-wide h tile, 512-token chunk): 128 WMMA steps, partial
// tiles merged with global_atomic_add_f32.  A built in registers from idx.
// ---------------------------------------------------------------------------
__global__ __launch_bounds__(32)
void k_dispatch(const float* __restrict__ x, const int* __restrict__ idxb,
                float* __restrict__ tok) {
    const int hbase = blockIdx.x * 16;
    const int tchunk = blockIdx.y * 512;
    const int lane = threadIdx.x;
    const int col = lane & 15;            // slot (A row) / h col (B,N) index
    const int hi = lane >> 4;             // half-wave: K offset 0 or 2
    const int e = col >> 1;
    const int ks = col & 1;

    v8f acc = {};
    for (int t = 0; t < 512; t += 4) {
        const int ta = tchunk + t + hi * 2;   // token for component .x
        v2f a, b;
        a.x = (idxb[ta * KK + ks] == e) ? 1.0f : 0.0f;
        a.y = (idxb[(ta + 1) * KK + ks] == e) ? 1.0f : 0.0f;
        b.x = x[(size_t)ta * HH + hbase + col];
        b.y = x[(size_t)(ta + 1) * HH + hbase + col];
        acc = wmma_f32_k4(a, b, acc);
    }
#pragma unroll
    for (int j = 0; j < 8; ++j) {
        const int r = j + hi * 8;             // D row = vgpr + 8*half (ISA 7.12.2)
        atomicAdd(&tok[r * HH + hbase + col], acc[j]);
    }
}

// ---------------------------------------------------------------------------
// K4: up-projection  midacc(e, 2 x F) += tok(e, 2 x H) * W1[e](H x F).
// Double-buffered async pipeline: 64x16 fp32 weight tiles (4KB) staged
// global->LDS with GLOBAL_LOAD_ASYNC_TO_LDS_B128 (8 instr x 512B per fill),
// s_wait_asynccnt<=8 gates the older buffer, WMMA consumes from LDS.
// H reduction split 4-way (16384 waves); W1 bytes touched exactly once.
// ---------------------------------------------------------------------------
__global__ __launch_bounds__(32)
void k_ffn1(const float* __restrict__ tok, const float* __restrict__ W1,
            float* __restrict__ midacc) {
    __shared__ __align__(16) float lds[2][64 * 16];
    const int fbase = blockIdx.x * 16;
    const int e = blockIdx.y;
    const int h0 = blockIdx.z * (HH / 4);       // 512 k-rows per block
    const int lane = threadIdx.x;
    const int col = lane & 15;
    const int hi = lane >> 4;
    const float* w = W1 + (size_t)e * HH * FF + fbase;
    const int trow = lane >> 2;                 // 0..7  (8 rows per issue)
    const int tc4 = (lane & 3) * 4;             // 16B column slice

    auto fill = [&](int buf, int kb0) {
#pragma unroll
        for (int i = 0; i < 8; ++i) {
            const int r = i * 8 + trow;
            stage16(w + (size_t)(kb0 + r) * FF + tc4, &lds[buf][r * 16 + tc4]);
        }
    };

    constexpr int NT = (HH / 4) / 64;           // 8 tiles
    fill(0, h0);
    fill(1, h0 + 64);

    v8f acc = {};
    for (int t = 0; t < NT; ++t) {
        if (t == NT - 1) wait_async_0(); else wait_async_8();
        const float* lb = lds[t & 1];
        const int kg0 = h0 + t * 64;
#pragma unroll 4
        for (int kk = 0; kk < 64; kk += 4) {
            const int kg = kg0 + kk + hi * 2;
            v2f a, b;
            a.x = (col < KK) ? tok[(e * KK + col) * HH + kg] : 0.0f;
            a.y = (col < KK) ? tok[(e * KK + col) * HH + kg + 1] : 0.0f;
            b.x = lb[(kk + hi * 2) * 16 + col];
            b.y = lb[(kk + hi * 2 + 1) * 16 + col];
            acc = wmma_f32_k4(a, b, acc);
        }
        if (t + 2 < NT) fill(t & 1, h0 + (t + 2) * 64);
    }
    if (hi == 0) {  // only D rows 0,1 (the two real token-slots) are valid
        atomicAdd(&midacc[(e * KK + 0) * FF + fbase + col], acc[0]);
        atomicAdd(&midacc[(e * KK + 1) * FF + fbase + col], acc[1]);
    }
}

// ---------------------------------------------------------------------------
// K5: mid = gelu_exact(midacc + b1)   (512 KB pass, negligible)
// ---------------------------------------------------------------------------
__global__ __launch_bounds__(256)
void k_act(const float* __restrict__ midacc, const float* __restrict__ b1,
           float* __restrict__ mid) {
    const int i = blockIdx.x * 256 + threadIdx.x;       // [0, 16*FF)
    const int r = i >> 13;                // slot
    const int f = i & (FF - 1);
    mid[i] = gelu_exact(midacc[i] + b1[(r >> 1) * FF + f]);
}

// ---------------------------------------------------------------------------
// K6: down-projection  Eo(e, 2 x H) += mid(e, 2 x F) * W2[e](F x H).
// Same async double-buffered pipeline; F reduction split 8-way (8192 waves).
// Eo pre-seeded with b2 so atomics land on the biased value.
// ---------------------------------------------------------------------------
__global__ __launch_bounds__(32)
void k_ffn2(const float* __restrict__ mid, const float* __restrict__ W2,
            float* __restrict__ Eo) {
    __shared__ __align__(16) float lds[2][64 * 16];
    const int hbase = blockIdx.x * 16;
    const int e = blockIdx.y;
    const int f0 = blockIdx.z * (FF / 8);       // 1024 k-rows per block
    const int lane = threadIdx.x;
    const int col = lane & 15;
    const int hi = lane >> 4;
    const float* w = W2 + (size_t)e * FF * HH + hbase;
    const int trow = lane >> 2;
    const int tc4 = (lane & 3) * 4;

    auto fill = [&](int buf, int kb0) {
#pragma unroll
        for (int i = 0; i < 8; ++i) {
            const int r = i * 8 + trow;
            stage16(w + (size_t)(kb0 + r) * HH + tc4, &lds[buf][r * 16 + tc4]);
        }
    };

    constexpr int NT = (FF / 8) / 64;           // 16 tiles
    fill(0, f0);
    fill(1, f0 + 64);

    v8f acc = {};
    for (int t = 0; t < NT; ++t) {
        if (t == NT - 1) wait_async_0(); else wait_async_8();
        const float* lb = lds[t & 1];
        const int kg0 = f0 + t * 64;
#pragma unroll 4
        for (int kk = 0; kk < 64; kk += 4) {
            const int kg = kg0 + kk + hi * 2;
            v2f a, b;
            a.x = (col < KK) ? mid[(e * KK + col) * FF + kg] : 0.0f;
            a.y = (col < KK) ? mid[(e * KK + col) * FF + kg + 1] : 0.0f;
            b.x = lb[(kk + hi * 2) * 16 + col];
            b.y = lb[(kk + hi * 2 + 1) * 16 + col];
            acc = wmma_f32_k4(a, b, acc);
        }
        if (t + 2 < NT) fill(t & 1, f0 + (t + 2) * 64);
    }
    if (hi == 0) {
        atomicAdd(&Eo[(e * KK + 0) * HH + hbase + col], acc[0]);
        atomicAdd(&Eo[(e * KK + 1) * HH + hbase + col], acc[1]);
    }
}

// ---------------------------------------------------------------------------
// K7: S(k, h) = sum_e Eo(e, k, h)       (the combine never uses idx!)
// ---------------------------------------------------------------------------
__global__ __launch_bounds__(256)
void k_sumexp(const float* __restrict__ Eo, float* __restrict__ S) {
    const int i = blockIdx.x * 256 + threadIdx.x;       // [0, KK*HH)
    const int k = i >> 11;
    const int h = i & (HH - 1);
    float s = 0.0f;
#pragma unroll
    for (int e = 0; e < NE; ++e) s += Eo[(e * KK + k) * HH + h];
    S[i] = s;
}

// ---------------------------------------------------------------------------
// K8: y(n, h) = G(n,0)*S(0,h) + G(n,1)*S(1,h).  Store-bound (64 MB),
// float4-vectorized, S is L2-resident.
// ---------------------------------------------------------------------------
__global__ __launch_bounds__(256)
void k_combine(const float* __restrict__ G, const float* __restrict__ S,
               float* __restrict__ y) {
    const size_t q = (size_t)blockIdx.x * 256 + threadIdx.x;  // float4 index
    const size_t i = q * 4;
    const int n = (int)(i >> 11);             // HH = 2048, so all 4 share n
    const int h = (int)(i & (HH - 1));
    const float g0 = G[n * KK + 0];
    const float g1 = G[n * KK + 1];
    const float4 s0 = *(const float4*)(S + h);
    const float4 s1 = *(const float4*)(S + HH + h);
    float4 r;
    r.x = fmaf(g0, s0.x, g1 * s1.x);
    r.y = fmaf(g0, s0.y, g1 * s1.y);
    r.z = fmaf(g0, s0.z, g1 * s1.z);
    r.w = fmaf(g0, s0.w, g1 * s1.w);
    *(float4*)(y + i) = r;
}

// ---------------------------------------------------------------------------
extern "C" void kernel_launch(void* const* d_in, const int* in_sizes, int n_in,
                              void* d_out, int out_size, void* d_ws, size_t ws_size,
                              hipStream_t stream) {
    const float* x  = (const float*)d_in[0];
    const float* Wr = (const float*)d_in[1];
    const float* br = (const float*)d_in[2];
    const float* W1 = (const float*)d_in[3];
    const float* b1 = (const float*)d_in[4];
    const float* W2 = (const float*)d_in[5];
    const float* b2 = (const float*)d_in[6];
    float* y = (float*)d_out;

    // workspace carve-out (~1.4 MB total)
    float* G      = (float*)d_ws;                       // NTOK*KK
    int*   idxb   = (int*)(G + NTOK * KK);              // NTOK*KK
    float* tok    = (float*)(idxb + NTOK * KK);         // NSLOT*HH
    float* midacc = tok + NSLOT * HH;                   // NSLOT*FF
    float* mid    = midacc + NSLOT * FF;                // NSLOT*FF
    float* Eo     = mid + NSLOT * FF;                   // NSLOT*HH
    float* Sbuf   = Eo + NSLOT * HH;                    // KK*HH

    // 1. router (8192 waves)
    k_router<<<NTOK / 8, 256, 0, stream>>>(x, Wr, br, G, idxb);
    // 2. init scratch
    k_init<<<(NSLOT * FF) / 256, 256, 0, stream>>>(tok, midacc, Eo, b2);
    // 3. dispatch GEMM: tok = P^T * X   (2048 waves x 128 WMMA)
    k_dispatch<<<dim3(HH / 16, NTOK / 512), 32, 0, stream>>>(x, idxb, tok);
    // 4. up-proj: stream 512 MB of W1 once via async-LDS double buffering
    k_ffn1<<<dim3(FF / 16, NE, 4), 32, 0, stream>>>(tok, W1, midacc);
    // 5. bias + exact GELU
    k_act<<<(NSLOT * FF) / 256, 256, 0, stream>>>(midacc, b1, mid);
    // 6. down-proj: stream 512 MB of W2 once via async-LDS double buffering
    k_ffn2<<<dim3(HH / 16, NE, 8), 32, 0, stream>>>(mid, W2, Eo);
    // 7. sum over experts
    k_sumexp<<<(KK * HH) / 256, 256, 0, stream>>>(Eo, Sbuf);
    // 8. combine: y = G * S   (64 MB store-bound)
    k_combine<<<(NTOK * HH / 4) / 256, 256, 0, stream>>>(G, Sbuf, y);
}